// Controller_35046933135834
// MI455X (gfx1250) — compile-verified
//
#include <hip/hip_runtime.h>
#include <math.h>

typedef __attribute__((ext_vector_type(2))) float v2f;
typedef __attribute__((ext_vector_type(4))) float v4f;
typedef __attribute__((ext_vector_type(8))) float v8f;

#define N_AGENTS 256
#define NFULL    1024
#define BATCH    4096
#define TSTEPS   8
#define LOG2F_   0.69314718055994530942f
#define TILES_PER_WAVE 4

// Workspace float offsets
#define WS_K1   0
#define WS_K2   4096
#define WS_K1T  8192
#define WS_K2T  12288
#define WS_B1   16384
#define WS_B2   17408
#define WS_L    18432
#define WS_U    22528
#define WS_D4   26624

// ---------------------------------------------------------------------------
// Prep: gather block-diagonal 4x4 K slices at t_idx, their transposes, biases,
// and ring off-diagonal blocks of A. One thread per agent.
// ---------------------------------------------------------------------------
__global__ void prep_kernel(const float* __restrict__ K, const float* __restrict__ b,
                            const float* __restrict__ GG, const int* __restrict__ tptr,
                            float* __restrict__ ws) {
    int a = threadIdx.x;
    int tv = tptr[0];
    int t_idx = (int)((double)tv / 0.1);
    if (t_idx > TSTEPS - 1) t_idx = TSTEPS - 1;
    if (t_idx < 0) t_idx = 0;
    const size_t LSTR = (size_t)NFULL * NFULL * TSTEPS;   // layer stride in K
    const size_t RSTR = (size_t)NFULL * TSTEPS;           // row stride in K

    for (int r = 0; r < 4; ++r) {
        for (int c = 0; c < 4; ++c) {
            size_t row = (size_t)(4 * a + r), col = (size_t)(4 * a + c);
            float k1 = K[row * RSTR + col * TSTEPS + t_idx];
            float k2 = K[LSTR + row * RSTR + col * TSTEPS + t_idx];
            ws[WS_K1  + a * 16 + r * 4 + c] = k1;
            ws[WS_K2  + a * 16 + r * 4 + c] = k2;
            ws[WS_K1T + a * 16 + c * 4 + r] = k1;
            ws[WS_K2T + a * 16 + c * 4 + r] = k2;
        }
        ws[WS_B1 + a * 4 + r] = b[(size_t)(4 * a + r) * TSTEPS + t_idx];
        ws[WS_B2 + a * 4 + r] = b[(size_t)NFULL * TSTEPS + (size_t)(4 * a + r) * TSTEPS + t_idx];
    }
    int am = (a + N_AGENTS - 1) & (N_AGENTS - 1);
    int ap = (a + 1) & (N_AGENTS - 1);
    for (int r = 0; r < 4; ++r)
        for (int c = 0; c < 4; ++c) {
            ws[WS_L + a * 16 + r * 4 + c] =
                GG[(size_t)(4 * a + r) * NFULL + 4 * am + c] -
                GG[(size_t)(4 * am + c) * NFULL + 4 * a + r];
            ws[WS_U + a * 16 + r * 4 + c] =
                GG[(size_t)(4 * a + r) * NFULL + 4 * ap + c] -
                GG[(size_t)(4 * ap + c) * NFULL + 4 * a + r];
        }
    if (a == 0) {
        // D4 = J4 - R4
        const float d4[16] = { -1.f, 0.f, -1.f, 0.f,
                                0.f, -1.f, 0.f, -1.f,
                                1.f,  0.f, 0.f,  0.f,
                                0.f,  1.f, 0.f,  0.f };
        for (int i = 0; i < 16; ++i) ws[WS_D4 + i] = d4[i];
    }
}

// ---------------------------------------------------------------------------
// Branch-free transcendentals on hardware TRANS units.
// ---------------------------------------------------------------------------
__device__ __forceinline__ float fast_tanh(float u) {
#if __has_builtin(__builtin_amdgcn_tanhf)
    return __builtin_amdgcn_tanhf(u);           // v_tanh_f32
#elif __has_builtin(__builtin_amdgcn_tanh_f32)
    return __builtin_amdgcn_tanh_f32(u);
#else
    float e = __expf(-2.0f * fabsf(u));         // v_exp_f32
    float t = (1.0f - e) * __builtin_amdgcn_rcpf(1.0f + e);  // v_rcp_f32
    return copysignf(t, u);
#endif
}

// Fused z = logcosh(u) and t = tanh(u): one exp, one log, one rcp.
__device__ __forceinline__ float logcosh_grad_z(float u, float& t_out) {
    float aa = fabsf(u);
    float e  = __expf(-2.0f * aa);                            // v_exp_f32
    float t  = (1.0f - e) * __builtin_amdgcn_rcpf(1.0f + e);  // v_rcp_f32
    t_out = copysignf(t, u);
    return aa + __logf(1.0f + e) - LOG2F_;                    // v_log_f32
}

// ---------------------------------------------------------------------------
// WMMA helpers (V_WMMA_F32_16X16X4_F32): A 16x4 (2 VGPR), B 4x16 (2 VGPR),
// C/D 16x16 (8 VGPR). A rows 0-3 and 8-11 carry the 4x4 matrix (duplicated),
// so feature r of the result lives in D VGPR r with lanes 16-31 duplicates.
// ---------------------------------------------------------------------------
__device__ __forceinline__ v8f wmma4(v2f A, v2f B, v8f C) {
    return __builtin_amdgcn_wmma_f32_16x16x4_f32(false, A, false, B, (short)0, C,
                                                 false, false);
}

// A-matrix lane pattern: lane<16: v0=A[r,0],v1=A[r,1]; lane>=16: v0=A[r,2],v1=A[r,3]
__device__ __forceinline__ v2f loadA(const float* __restrict__ M, int rr, int col0) {
    v2f a; a.x = 0.f; a.y = 0.f;
    if (rr >= 0) { a.x = M[rr * 4 + col0]; a.y = M[rr * 4 + col0 + 1]; }
    return a;
}

// Feature-VGPR layout (with hi-lane duplicates) -> B 4x16 layout:
// v0: row0 (lanes 0-15) / row2 (lanes 16-31); v1: row1 / row3.
__device__ __forceinline__ v2f toB(v8f d, bool lo) {
    v2f b;
    b.x = lo ? d[0] : d[2];
    b.y = lo ? d[1] : d[3];
    return b;
}

struct AgentMats { v2f A1, A2, A1T, A2T; v8f CB1, CB2; };

__device__ __forceinline__ AgentMats loadAgent(const float* __restrict__ ws, int a,
                                               int rr, int col0) {
    AgentMats m;
    m.A1  = loadA(ws + WS_K1  + a * 16, rr, col0);
    m.A2  = loadA(ws + WS_K2  + a * 16, rr, col0);
    m.A1T = loadA(ws + WS_K1T + a * 16, rr, col0);
    m.A2T = loadA(ws + WS_K2T + a * 16, rr, col0);
    const float* b1 = ws + WS_B1 + a * 4;
    const float* b2 = ws + WS_B2 + a * 4;
    v8f c1 = {}, c2 = {};
#pragma unroll
    for (int r = 0; r < 4; ++r) { c1[r] = b1[r]; c2[r] = b2[r]; }
    m.CB1 = c1; m.CB2 = c2;
    return m;
}

// Per-agent gradient chain for a 16-batch tile:
// g = K1^T ( tanh(u1) * ( K2^T tanh(K2 logcosh(u1) + b2) ) ),  u1 = K1 x + b1
// Returns g in B-matrix (4x16) layout ready for the output WMMAs.
__device__ __forceinline__ v2f grad_chain(const float* __restrict__ x, int a, int base,
                                          bool lo, int n, int k0, const AgentMats& m) {
    const float* xp = x + (size_t)(4 * a + k0) * BATCH + base + n;
    v2f bx; bx.x = xp[0]; bx.y = xp[BATCH];

    v8f u1 = wmma4(m.A1, bx, m.CB1);
    v8f z1 = {}; float t1[4];
#pragma unroll
    for (int r = 0; r < 4; ++r) z1[r] = logcosh_grad_z(u1[r], t1[r]);

    v8f u2 = wmma4(m.A2, toB(z1, lo), m.CB2);
    v8f gu2 = {};
#pragma unroll
    for (int r = 0; r < 4; ++r) gu2[r] = fast_tanh(u2[r]);

    v8f zero = {};
    v8f gz1 = wmma4(m.A2T, toB(gu2, lo), zero);
    v8f gu1 = {};
#pragma unroll
    for (int r = 0; r < 4; ++r) gu1[r] = t1[r] * gz1[r];

    v8f g = wmma4(m.A1T, toB(gu1, lo), zero);
    return toB(g, lo);
}

// ---------------------------------------------------------------------------
// Main kernel: 8 waves / block, one agent per wave, TILES_PER_WAVE 16-batch
// tiles per wave. out_block(a) = D4*g_a + L_a*g_{a-1} + U_a*g_{a+1}.
// ---------------------------------------------------------------------------
__global__ void __launch_bounds__(256)
ctrl_kernel(const float* __restrict__ x, const float* __restrict__ ws,
            float* __restrict__ out) {
    int lane = threadIdx.x & 31;
    int wave = threadIdx.x >> 5;
    int a  = blockIdx.y * 8 + wave;
    int am = (a + N_AGENTS - 1) & (N_AGENTS - 1);
    int ap = (a + 1) & (N_AGENTS - 1);

    int r = lane & 15, hi = lane >> 4;
    bool lo = (hi == 0);
    int rr = (r < 4) ? r : ((r >= 8 && r < 12) ? (r - 8) : -1);
    int col0 = hi * 2;
    int n = r;       // batch column within tile (B/C/D lane mapping)
    int k0 = col0;   // feature pair selector for B loads

    AgentMats Mm = loadAgent(ws, am, rr, col0);
    AgentMats M0 = loadAgent(ws, a,  rr, col0);
    AgentMats Mp = loadAgent(ws, ap, rr, col0);
    v2f AL = loadA(ws + WS_L + a * 16, rr, col0);
    v2f AU = loadA(ws + WS_U + a * 16, rr, col0);
    v2f AD = loadA(ws + WS_D4,         rr, col0);

    int base0 = blockIdx.x * (16 * TILES_PER_WAVE);
#pragma unroll
    for (int tI = 0; tI < TILES_PER_WAVE; ++tI) {
        int base = base0 + tI * 16;
        v2f gm = grad_chain(x, am, base, lo, n, k0, Mm);
        v2f g0 = grad_chain(x, a,  base, lo, n, k0, M0);
        v2f gp = grad_chain(x, ap, base, lo, n, k0, Mp);

        v8f zero = {};
        v8f acc = wmma4(AL, gm, zero);
        acc = wmma4(AU, gp, acc);
        acc = wmma4(AD, g0, acc);

        if (lo) {
            v4f o; o.x = acc[0]; o.y = acc[1]; o.z = acc[2]; o.w = acc[3];
            *(v4f*)(out + (size_t)(base + n) * NFULL + 4 * a) = o;
        }
    }
}

extern "C" void kernel_launch(void* const* d_in, const int* in_sizes, int n_in,
                              void* d_out, int out_size, void* d_ws, size_t ws_size,
                              hipStream_t stream) {
    const float* x  = (const float*)d_in[0];
    const float* K  = (const float*)d_in[1];
    const float* b  = (const float*)d_in[2];
    const float* GG = (const float*)d_in[3];
    const int*   t  = (const int*)d_in[4];
    float* out = (float*)d_out;
    float* ws  = (float*)d_ws;

    prep_kernel<<<1, N_AGENTS, 0, stream>>>(K, b, GG, t, ws);

    dim3 grid(BATCH / (16 * TILES_PER_WAVE), N_AGENTS / 8);
    ctrl_kernel<<<grid, 256, 0, stream>>>(x, ws, out);
}